// DWSA_Loss_34196529611297
// MI455X (gfx1250) — compile-verified
//
#include <hip/hip_runtime.h>
#include <math.h>

#define LA    4096
#define LB    4096
#define KDIM  1024
#define NCOL  (2 * LB + 1)          // 8193
#define THRESH 2.0f
#define GAMMA  0.01f
#define INV_GAMMA (1.0f / GAMMA)
#define EPSN   1e-10f

typedef __attribute__((ext_vector_type(2))) float v2f;
typedef __attribute__((ext_vector_type(8))) float v8f;

// numerically stable logaddexp with -inf identity
__device__ __forceinline__ float lae(float a, float b) {
  float m = fmaxf(a, b);
  if (m == -INFINITY) return -INFINITY;
  return m + log1pf(expf(fminf(a, b) - m));
}

// ---------------------------------------------------------------------------
// Phase 1: L2-normalize rows (one block per row)
// ---------------------------------------------------------------------------
__global__ void __launch_bounds__(256) normalize_rows(const float* __restrict__ src,
                                                      float* __restrict__ dst) {
  __shared__ float red[256];
  const int row = blockIdx.x;
  const float* s = src + (size_t)row * KDIM;
  float acc = 0.0f;
  for (int c = threadIdx.x; c < KDIM; c += 256) { float v = s[c]; acc += v * v; }
  red[threadIdx.x] = acc;
  __syncthreads();
  for (int off = 128; off > 0; off >>= 1) {
    if (threadIdx.x < off) red[threadIdx.x] += red[threadIdx.x + off];
    __syncthreads();
  }
  const float inv = rsqrtf(red[0] + EPSN);
  float* d = dst + (size_t)row * KDIM;
  for (int c = threadIdx.x; c < KDIM; c += 256) d[c] = s[c] * inv;
}

// ---------------------------------------------------------------------------
// Phase 2: matching = 1 - An * Bn^T via fp32 WMMA (16x16x4), 32x32 tile/wave
// A-16x4 fp32 layout: lane l holds row M=l%16, K = 2*(l/16) + component.
// B (4x16) symmetric over N. C/D: VGPR v, lanes 0-15 -> M=v, 16-31 -> M=v+8.
// ---------------------------------------------------------------------------
__global__ void __launch_bounds__(256) cosine_gemm_wmma(const float* __restrict__ An,
                                                        const float* __restrict__ Bn,
                                                        float* __restrict__ Mt) {
  const int lane  = threadIdx.x & 31;
  const int wave  = blockIdx.x * (blockDim.x >> 5) + (threadIdx.x >> 5);
  const int tx    = wave & 127;          // 128 tiles of 32 cols
  const int ty    = wave >> 7;           // 128 tiles of 32 rows
  const int row0  = ty * 32;
  const int col0  = tx * 32;
  const int lrow  = lane & 15;
  const int khalf = lane >> 4;

  v8f acc00 = {0,0,0,0,0,0,0,0};
  v8f acc01 = {0,0,0,0,0,0,0,0};
  v8f acc10 = {0,0,0,0,0,0,0,0};
  v8f acc11 = {0,0,0,0,0,0,0,0};

  const float* pa0 = An + (size_t)(row0 + lrow)      * KDIM + 2 * khalf;
  const float* pa1 = An + (size_t)(row0 + 16 + lrow) * KDIM + 2 * khalf;
  const float* pb0 = Bn + (size_t)(col0 + lrow)      * KDIM + 2 * khalf;
  const float* pb1 = Bn + (size_t)(col0 + 16 + lrow) * KDIM + 2 * khalf;

#pragma unroll 4
  for (int kb = 0; kb < KDIM; kb += 4) {
    v2f a0 = *(const v2f*)(pa0 + kb);
    v2f a1 = *(const v2f*)(pa1 + kb);
    v2f b0 = *(const v2f*)(pb0 + kb);
    v2f b1 = *(const v2f*)(pb1 + kb);
    acc00 = __builtin_amdgcn_wmma_f32_16x16x4_f32(false, a0, false, b0, (short)0, acc00, false, false);
    acc01 = __builtin_amdgcn_wmma_f32_16x16x4_f32(false, a0, false, b1, (short)0, acc01, false, false);
    acc10 = __builtin_amdgcn_wmma_f32_16x16x4_f32(false, a1, false, b0, (short)0, acc10, false, false);
    acc11 = __builtin_amdgcn_wmma_f32_16x16x4_f32(false, a1, false, b1, (short)0, acc11, false, false);
  }

  const int m0 = row0 + 8 * khalf;
  const int n0 = col0 + lrow;
#pragma unroll
  for (int v = 0; v < 8; ++v) {
    Mt[(size_t)(m0 + v)      * LB + n0]        = 1.0f - acc00[v];
    Mt[(size_t)(m0 + v)      * LB + (n0 + 16)] = 1.0f - acc01[v];
    Mt[(size_t)(m0 + 16 + v) * LB + n0]        = 1.0f - acc10[v];
    Mt[(size_t)(m0 + 16 + v) * LB + (n0 + 16)] = 1.0f - acc11[v];
  }
}

// ---------------------------------------------------------------------------
// Phase 3: per-row softmax stats of the interleaved cost row
//   row i values: {T} ∪ {matching[i,j], T}_j  -> (LB+1) threshold entries
// ---------------------------------------------------------------------------
__global__ void __launch_bounds__(256) row_stats(const float* __restrict__ Mt,
                                                 float* __restrict__ rmax,
                                                 float* __restrict__ rsum) {
  __shared__ float red[256];
  const int row = blockIdx.x;
  const float* s = Mt + (size_t)row * LB;

  float mx = THRESH;
  for (int c = threadIdx.x; c < LB; c += 256) mx = fmaxf(mx, s[c]);
  red[threadIdx.x] = mx;
  __syncthreads();
  for (int off = 128; off > 0; off >>= 1) {
    if (threadIdx.x < off) red[threadIdx.x] = fmaxf(red[threadIdx.x], red[threadIdx.x + off]);
    __syncthreads();
  }
  mx = red[0];
  __syncthreads();

  float sum = 0.0f;
  for (int c = threadIdx.x; c < LB; c += 256) sum += expf(s[c] - mx);
  red[threadIdx.x] = sum;
  __syncthreads();
  for (int off = 128; off > 0; off >>= 1) {
    if (threadIdx.x < off) red[threadIdx.x] += red[threadIdx.x + off];
    __syncthreads();
  }
  if (threadIdx.x == 0) {
    rmax[row] = mx;
    rsum[row] = red[0] + (float)(LB + 1) * expf(THRESH - mx);
  }
}

// ---------------------------------------------------------------------------
// Phase 4: sequential soft-DTW DP. One 1024-thread block (32 wave32s).
// D row lives in LDS; per step: prefix logaddexp of -prev/gamma.
// chunk start is always even -> sel[j]=j-(j&1) never leaves the chunk.
// ---------------------------------------------------------------------------
__global__ void __launch_bounds__(1024) dwsa_dp(const float* __restrict__ Mt,
                                                const float* __restrict__ rmax,
                                                const float* __restrict__ rsum,
                                                float* __restrict__ out) {
  __shared__ float prev[NCOL];
  __shared__ float wtot[32];
  const int tid  = threadIdx.x;
  const int lane = tid & 31;
  const int wid  = tid >> 5;
  const int beg  = tid * 8;
  const int cnt  = (tid == 1023) ? 9 : 8;   // thread 1023 also owns column 8192

  // D[0] = C[0] (softmaxed row 0)
  {
    const float mx = rmax[0];
    const float isum = 1.0f / rsum[0];
    const float cT = expf(THRESH - mx) * isum;
    for (int k = 0; k < cnt; ++k) {
      const int j = beg + k;
      prev[j] = (j & 1) ? expf(Mt[(size_t)((j - 1) >> 1)] - mx) * isum : cT;
    }
  }
  __syncthreads();

  float loc[9];
  for (int i = 1; i < LA; ++i) {
    const float mx   = rmax[i];
    const float isum = 1.0f / rsum[i];
    const float cT   = expf(THRESH - mx) * isum;
    const float* mrow = Mt + (size_t)i * LB;

    // (a) per-thread inclusive logaddexp scan of x = -prev/gamma
    float run = -INFINITY;
    for (int k = 0; k < cnt; ++k) {
      run = lae(run, -prev[beg + k] * INV_GAMMA);
      loc[k] = run;
    }
    // (b) wave32 inclusive scan of thread totals
    float v = run;
    for (int off = 1; off < 32; off <<= 1) {
      float u = __shfl_up(v, off, 32);
      if (lane >= off) v = lae(v, u);
    }
    float up = __shfl_up(v, 1, 32);
    const float excl_lane = (lane == 0) ? -INFINITY : up;

    __syncthreads();                 // protect wtot reuse across iterations
    if (lane == 31) wtot[wid] = v;
    __syncthreads();
    if (wid == 0) {                  // (c) scan the 32 wave totals
      float w = wtot[lane];
      for (int off = 1; off < 32; off <<= 1) {
        float u = __shfl_up(w, off, 32);
        if (lane >= off) w = lae(w, u);
      }
      wtot[lane] = w;
    }
    __syncthreads();
    const float base = (wid == 0) ? excl_lane : lae(wtot[wid - 1], excl_lane);

    // (d) new row: D[j] = C[i,j] - gamma * LSE[sel[j]]
    for (int k = 0; k < cnt; ++k) {
      const int j  = beg + k;
      const int ks = k - (j & 1);
      const float L  = lae(base, loc[ks]);
      const float cv = (j & 1) ? expf(mrow[(j - 1) >> 1] - mx) * isum : cT;
      prev[j] = cv - GAMMA * L;
    }
    if (i + 1 < LA)   // pull next matching row toward the WGP (global_prefetch_b8)
      __builtin_prefetch(Mt + (size_t)(i + 1) * LB + (beg >> 1), 0, 1);
  }

  // final: softmin = -gamma * LSE(-last/gamma); loss = softmin / La
  float run = -INFINITY;
  for (int k = 0; k < cnt; ++k) run = lae(run, -prev[beg + k] * INV_GAMMA);
  for (int off = 1; off < 32; off <<= 1) {
    float u = __shfl_up(run, off, 32);
    if (lane >= off) run = lae(run, u);
  }
  __syncthreads();
  if (lane == 31) wtot[wid] = run;
  __syncthreads();
  if (tid == 0) {
    float t = wtot[0];
    for (int w = 1; w < 32; ++w) t = lae(t, wtot[w]);
    out[0] = (-GAMMA * t) / (float)LA;
  }
}

// ---------------------------------------------------------------------------
extern "C" void kernel_launch(void* const* d_in, const int* in_sizes, int n_in,
                              void* d_out, int out_size, void* d_ws, size_t ws_size,
                              hipStream_t stream) {
  const float* ca = (const float*)d_in[0];
  const float* cb = (const float*)d_in[1];
  float* ws   = (float*)d_ws;
  float* An   = ws;                               // 4096*1024
  float* Bn   = An + (size_t)LA * KDIM;           // 4096*1024
  float* Mt   = Bn + (size_t)LB * KDIM;           // 4096*4096
  float* rmax = Mt + (size_t)LA * LB;             // 4096
  float* rsum = rmax + LA;                        // 4096
  float* outp = (float*)d_out;

  normalize_rows<<<LA, 256, 0, stream>>>(ca, An);
  normalize_rows<<<LB, 256, 0, stream>>>(cb, Bn);
  cosine_gemm_wmma<<<(128 * 128) / 8, 256, 0, stream>>>(An, Bn, Mt);
  row_stats<<<LA, 256, 0, stream>>>(Mt, rmax, rsum);
  dwsa_dp<<<1, 1024, 0, stream>>>(Mt, rmax, rsum, outp);
}